// VoxelResBackBone8xVoxelNeXtSPS_39479339385416
// MI455X (gfx1250) — compile-verified
//
#include <hip/hip_runtime.h>

typedef __attribute__((ext_vector_type(2))) float v2f;
typedef __attribute__((ext_vector_type(8))) float v8f;

#define BN_EPS 1e-3f

// ---------------------------------------------------------------------------
// utility fills
// ---------------------------------------------------------------------------
__global__ void zerof_k(float* __restrict__ p, long n) {
  long i = (long)blockIdx.x * blockDim.x + threadIdx.x;
  if (i < n) p[i] = 0.0f;
}

__global__ void zerou_k(unsigned* __restrict__ p, int n) {
  int i = blockIdx.x * blockDim.x + threadIdx.x;
  if (i < n) p[i] = 0u;
}

// ---------------------------------------------------------------------------
// Sparse conv: gather (rin) -> 16x16 tile GEMM via V_WMMA_F32_16X16X4_F32 ->
// scatter-add (rout). Grid: z = 27 kernel offsets, y = 16-col tile,
// x = pair-tile blocks (8 waves/block, persistent over pair tiles).
//
// A (16x4 f32):  lane L (0..15) holds A[M=L][K=0..1] in v2f,
//                lane L+16 holds A[M=L][K=2..3]                (ISA 7.12.2)
// B (4x16 f32):  lane L (0..15) holds B[K=0..1][N=L],
//                lane L+16 holds B[K=2..3][N=L]
// C/D (16x16):   VGPR j: lanes 0..15 -> D[M=j][N=lane],
//                        lanes 16..31 -> D[M=j+8][N=lane-16]
// Sentinel rows (index == n_in / n_out) gather zero / skip scatter.
// ---------------------------------------------------------------------------
template <int Cin, int Cout>
__global__ void spconv_wmma_k(const float* __restrict__ feat, int n_in,
                              const float* __restrict__ W,
                              const int* __restrict__ rin, const int* __restrict__ rout,
                              int P, int n_out, float* __restrict__ out, int ptiles) {
  constexpr int KS = (Cin + 3) / 4;  // K-steps of 4

  const int k    = blockIdx.z;        // kernel offset 0..26
  const int n0   = blockIdx.y << 4;   // output-column tile base
  const int lane = threadIdx.x & 31;
  const int m    = lane & 15;
  const int hi   = lane >> 4;

  // B fragments: invariant for this (k, column tile) — load once per wave.
  v2f b[KS];
#pragma unroll
  for (int s = 0; s < KS; ++s) {
    int c0 = 4 * s + hi * 2;
    int c1 = c0 + 1;
    b[s].x = (c0 < Cin) ? W[((long)k * Cin + c0) * Cout + n0 + m] : 0.0f;
    b[s].y = (c1 < Cin) ? W[((long)k * Cin + c1) * Cout + n0 + m] : 0.0f;
  }

  const int* __restrict__ rin_k  = rin + (long)k * P;
  const int* __restrict__ rout_k = rout + (long)k * P;

  const int wpb = blockDim.x >> 5;  // waves per block
  for (int pt = blockIdx.x * wpb + (threadIdx.x >> 5); pt < ptiles;
       pt += gridDim.x * wpb) {
    const int p0   = pt << 4;
    const int pidx = p0 + m;
    const int rowin = (pidx < P) ? rin_k[pidx] : n_in;
    const bool rv   = rowin < n_in;
    const float* frow = feat + (long)rowin * Cin;

    v8f acc = {};
#pragma unroll
    for (int s = 0; s < KS; ++s) {
      int c0 = 4 * s + hi * 2;
      int c1 = c0 + 1;
      v2f a;
      a.x = (rv && c0 < Cin) ? frow[c0] : 0.0f;
      a.y = (rv && c1 < Cin) ? frow[c1] : 0.0f;
      acc = __builtin_amdgcn_wmma_f32_16x16x4_f32(false, a, false, b[s],
                                                  (short)0, acc, false, false);
    }

#pragma unroll
    for (int j = 0; j < 8; ++j) {
      int po   = p0 + hi * 8 + j;  // pair index for D row M = j + hi*8
      int rowo = (po < P) ? rout_k[po] : n_out;
      if (rowo < n_out) atomicAdd(out + (long)rowo * Cout + n0 + m, acc[j]);
    }
  }
}

// ---------------------------------------------------------------------------
// BatchNorm (training mode): per-channel sum / sumsq -> scale/shift -> apply.
// stats layout (floats): [0..31]=sum [32..63]=sumsq [64..95]=scale [96..127]=shift
// ---------------------------------------------------------------------------
__global__ void bn_stats_k(const float* __restrict__ x, long total, int C,
                           float* __restrict__ stats) {
  int tid = threadIdx.x;
  float s = 0.0f, q = 0.0f;
  long stride = (long)gridDim.x * blockDim.x;  // multiple of C (C | 256)
  for (long i = (long)blockIdx.x * blockDim.x + tid; i < total; i += stride) {
    float v = x[i];
    s += v;
    q += v * v;
  }
  __shared__ float ss[256];
  __shared__ float sq[256];
  ss[tid] = s;
  sq[tid] = q;
  __syncthreads();
  for (int off = 128; off >= C; off >>= 1) {
    if (tid < off) { ss[tid] += ss[tid + off]; sq[tid] += sq[tid + off]; }
    __syncthreads();
  }
  if (tid < C) {
    atomicAdd(&stats[tid], ss[tid]);
    atomicAdd(&stats[32 + tid], sq[tid]);
  }
}

__global__ void bn_finalize_k(float* __restrict__ stats, const float* __restrict__ g,
                              const float* __restrict__ beta, long n, int C) {
  int tid = threadIdx.x;
  if (tid < C) {
    float inv_n = 1.0f / (float)n;
    float mean  = stats[tid] * inv_n;
    float var   = stats[32 + tid] * inv_n - mean * mean;
    float sc    = g[tid] * rsqrtf(var + BN_EPS);
    stats[64 + tid] = sc;
    stats[96 + tid] = beta[tid] - mean * sc;
  }
}

// mode 0: y = sc*x+sh ; mode 1: relu(sc*x+sh) ; mode 2: relu(sc*x+sh + res)
__global__ void bn_apply_k(const float* __restrict__ x, const float* __restrict__ res,
                           float* __restrict__ y, long total, int C,
                           const float* __restrict__ stats, int mode) {
  long i = (long)blockIdx.x * blockDim.x + threadIdx.x;
  if (i >= total) return;
  int c = (int)(i % C);
  float v = x[i] * stats[64 + c] + stats[96 + c];
  if (mode == 2) v += res[i];
  if (mode >= 1) v = fmaxf(v, 0.0f);
  y[i] = v;
}

// ---------------------------------------------------------------------------
// Importance conv (16 -> 1) : scalar dot per rulebook pair, then sigmoid.
// ---------------------------------------------------------------------------
__global__ void imp_conv_k(const float* __restrict__ x, const float* __restrict__ Wp,
                           const int* __restrict__ rin, const int* __restrict__ rout,
                           int P, int n, float* __restrict__ imp) {
  long i = (long)blockIdx.x * blockDim.x + threadIdx.x;
  if (i >= (long)27 * P) return;
  int k  = (int)(i / P);
  int ri = rin[i];
  int ro = rout[i];
  if (ri < n && ro < n) {
    const float* f = x + (long)ri * 16;
    const float* w = Wp + k * 16;
    float s = 0.0f;
#pragma unroll
    for (int c = 0; c < 16; ++c) s += f[c] * w[c];
    atomicAdd(&imp[ro], s);
  }
}

__global__ void sigmoid_k(float* __restrict__ p, int n) {
  int i = blockIdx.x * blockDim.x + threadIdx.x;
  if (i < n) p[i] = 1.0f / (1.0f + expf(-p[i]));
}

// ---------------------------------------------------------------------------
// Exact k-th-largest via 4-pass radix select on positive-float bit patterns.
// sel[0]=prefix, sel[1]=remaining k.
// ---------------------------------------------------------------------------
__global__ void sel_init_k(unsigned* __restrict__ sel, unsigned k) {
  if (threadIdx.x == 0 && blockIdx.x == 0) { sel[0] = 0u; sel[1] = k; }
}

__global__ void sel_hist_k(const float* __restrict__ imp, int n, unsigned shift,
                           const unsigned* __restrict__ sel, unsigned* __restrict__ hist) {
  unsigned prefix = sel[0];
  unsigned maskhi = (shift + 8 >= 32) ? 0u : (0xFFFFFFFFu << (shift + 8));
  int i = blockIdx.x * blockDim.x + threadIdx.x;
  if (i < n) {
    unsigned u = __float_as_uint(imp[i]);
    if ((u & maskhi) == (prefix & maskhi))
      atomicAdd(&hist[(u >> shift) & 255u], 1u);
  }
}

__global__ void sel_scan_k(const unsigned* __restrict__ hist, unsigned* __restrict__ sel,
                           unsigned shift, float* __restrict__ kth, int last) {
  if (threadIdx.x == 0 && blockIdx.x == 0) {
    unsigned k = sel[1];
    unsigned cum = 0;
    unsigned chosen = 0;
    unsigned newk = k;
    for (int b = 255; b >= 0; --b) {
      unsigned c = hist[b];
      if (cum + c >= k) { chosen = (unsigned)b; newk = k - cum; break; }
      cum += c;
    }
    sel[0] |= chosen << shift;
    sel[1] = newk;
    if (last) *kth = __uint_as_float(sel[0]);
  }
}

__global__ void prune_apply_k(float* __restrict__ x, const float* __restrict__ imp,
                              const float* __restrict__ kth, long total) {
  long i = (long)blockIdx.x * blockDim.x + threadIdx.x;
  if (i >= total) return;
  int r = (int)(i >> 4);  // C = 16
  float im = imp[r];
  x[i] = (im >= *kth) ? x[i] * im : 0.0f;
}

// ---------------------------------------------------------------------------
// host-side helpers
// ---------------------------------------------------------------------------
static inline int cdiv_i(long a, long b) { return (int)((a + b - 1) / b); }

static void zero_f(float* p, long n, hipStream_t s) {
  zerof_k<<<cdiv_i(n, 256), 256, 0, s>>>(p, n);
}

template <int Cin, int Cout>
static void run_conv(const float* feat, int n_in, const float* W,
                     const int* rin, const int* rout, int P, int n_out, float* out,
                     hipStream_t s) {
  zero_f(out, (long)n_out * Cout, s);
  int ptiles = (P + 15) / 16;
  int gx = cdiv_i(ptiles, 8);
  if (gx > 1024) gx = 1024;  // persistent waves beyond this
  dim3 grid(gx, Cout / 16, 27);
  spconv_wmma_k<Cin, Cout><<<grid, 256, 0, s>>>(feat, n_in, W, rin, rout, P, n_out,
                                                out, ptiles);
}

static void run_bn(const float* x, const float* res, float* y, long n, int C,
                   const float* g, const float* beta, float* stats, int mode,
                   hipStream_t s) {
  zero_f(stats, 128, s);
  bn_stats_k<<<512, 256, 0, s>>>(x, n * C, C, stats);
  bn_finalize_k<<<1, 32, 0, s>>>(stats, g, beta, n, C);
  long total = n * C;
  bn_apply_k<<<cdiv_i(total, 256), 256, 0, s>>>(x, res, y, total, C, stats, mode);
}

// ---------------------------------------------------------------------------
// kernel_launch
// ---------------------------------------------------------------------------
extern "C" void kernel_launch(void* const* d_in, const int* in_sizes, int n_in_args,
                              void* d_out, int out_size, void* d_ws, size_t ws_size,
                              hipStream_t stream) {
  (void)n_in_args; (void)ws_size;

  const float* vox  = (const float*)d_in[0];   // (N1,5)
  const float* W_in = (const float*)d_in[1];   // (27,5,16)
  const float* g_in = (const float*)d_in[2];   // (16)
  const float* b_in = (const float*)d_in[3];   // (16)
  const float* Wb1  = (const float*)d_in[4];   // (2,2,27,16,16)
  const float* gb1  = (const float*)d_in[6];   // (2,2,16)
  const float* beb1 = (const float*)d_in[7];   // (2,2,16)
  const float* Wp   = (const float*)d_in[8];   // (27,16,1)
  const float* Wd   = (const float*)d_in[9];   // (27,16,32)
  const float* g_d  = (const float*)d_in[10];  // (32)
  const float* b_d  = (const float*)d_in[11];  // (32)
  const float* Wb2  = (const float*)d_in[12];  // (2,2,27,32,32)
  const float* gb2  = (const float*)d_in[14];  // (2,2,32)
  const float* beb2 = (const float*)d_in[15];  // (2,2,32)
  const int* rb1_in  = (const int*)d_in[16];
  const int* rb1_out = (const int*)d_in[17];
  const int* rbd_in  = (const int*)d_in[18];
  const int* rbd_out = (const int*)d_in[19];
  const int* rb2_in  = (const int*)d_in[20];
  const int* rb2_out = (const int*)d_in[21];
  // d_in[5]/d_in[13] (conv biases) cancel under training-mode BN; d_in[22]=n2 via out_size.

  const int N1 = in_sizes[0] / 5;
  const int P1 = in_sizes[16] / 27;
  const int Pd = in_sizes[18] / 27;
  const int P2 = in_sizes[20] / 27;
  const int N2 = out_size / 32;

  // workspace carve-up
  float* xa = (float*)d_ws;                    // (N1,16) stage-1 features
  float* xb = xa + (size_t)N1 * 16;            // (N1,16) tmp
  float* xc = xb + (size_t)N1 * 16;            // (N1,16) tmp
  float* yb = xc + (size_t)N1 * 16;            // (N2,32) tmp
  float* yc = yb + (size_t)N2 * 32;            // (N2,32) tmp
  float* imp = yc + (size_t)N2 * 32;           // (N1)
  float* stats = imp + (size_t)N1;             // 128 floats
  unsigned* hist = (unsigned*)(stats + 128);   // 256
  unsigned* sel  = hist + 256;                 // 2
  float* kth = (float*)(sel + 2);              // 1
  float* ya = (float*)d_out;                   // (N2,32) stage-2 features

  // --- conv_input: SubMConv3d 5->16 + BN + ReLU ---
  run_conv<5, 16>(vox, N1, W_in, rb1_in, rb1_out, P1, N1, xa, stream);
  run_bn(xa, nullptr, xa, N1, 16, g_in, b_in, stats, 1, stream);

  // --- conv1: 2x SparseBasicBlock @16ch ---
  for (int i = 0; i < 2; ++i) {
    const float* w0 = Wb1 + (size_t)(i * 2 + 0) * 27 * 16 * 16;
    const float* w1 = Wb1 + (size_t)(i * 2 + 1) * 27 * 16 * 16;
    run_conv<16, 16>(xa, N1, w0, rb1_in, rb1_out, P1, N1, xb, stream);
    run_bn(xb, nullptr, xb, N1, 16, gb1 + (i * 2 + 0) * 16, beb1 + (i * 2 + 0) * 16,
           stats, 1, stream);
    run_conv<16, 16>(xb, N1, w1, rb1_in, rb1_out, P1, N1, xc, stream);
    run_bn(xc, xa, xa, N1, 16, gb1 + (i * 2 + 1) * 16, beb1 + (i * 2 + 1) * 16,
           stats, 2, stream);
  }

  // --- DynamicFocalPruning: importance conv -> sigmoid -> exact kth -> mask ---
  zero_f(imp, N1, stream);
  imp_conv_k<<<cdiv_i((long)27 * P1, 256), 256, 0, stream>>>(xa, Wp, rb1_in, rb1_out,
                                                             P1, N1, imp);
  sigmoid_k<<<cdiv_i(N1, 256), 256, 0, stream>>>(imp, N1);

  unsigned kkeep = (unsigned)(N1 - N1 / 2);
  sel_init_k<<<1, 32, 0, stream>>>(sel, kkeep);
  for (int pass = 0; pass < 4; ++pass) {
    unsigned shift = 24u - 8u * pass;
    zerou_k<<<1, 256, 0, stream>>>(hist, 256);
    sel_hist_k<<<cdiv_i(N1, 256), 256, 0, stream>>>(imp, N1, shift, sel, hist);
    sel_scan_k<<<1, 32, 0, stream>>>(hist, sel, shift, kth, pass == 3);
  }
  prune_apply_k<<<cdiv_i((long)N1 * 16, 256), 256, 0, stream>>>(xa, imp, kth,
                                                                (long)N1 * 16);

  // --- strided SparseConv3d 16->32 + BN + ReLU ---
  run_conv<16, 32>(xa, N1, Wd, rbd_in, rbd_out, Pd, N2, ya, stream);
  run_bn(ya, nullptr, ya, N2, 32, g_d, b_d, stats, 1, stream);

  // --- conv2: 2x SparseBasicBlock @32ch (accumulates in d_out) ---
  for (int i = 0; i < 2; ++i) {
    const float* w0 = Wb2 + (size_t)(i * 2 + 0) * 27 * 32 * 32;
    const float* w1 = Wb2 + (size_t)(i * 2 + 1) * 27 * 32 * 32;
    run_conv<32, 32>(ya, N2, w0, rb2_in, rb2_out, P2, N2, yb, stream);
    run_bn(yb, nullptr, yb, N2, 32, gb2 + (i * 2 + 0) * 32, beb2 + (i * 2 + 0) * 32,
           stats, 1, stream);
    run_conv<32, 32>(yb, N2, w1, rb2_in, rb2_out, P2, N2, yc, stream);
    run_bn(yc, ya, ya, N2, 32, gb2 + (i * 2 + 1) * 32, beb2 + (i * 2 + 1) * 32,
           stats, 2, stream);
  }
}